// CRF_31765578121292
// MI455X (gfx1250) — compile-verified
//
#include <hip/hip_runtime.h>

#define B_ 64
#define T_ 1024
#define M_ 128
#define MP 132   // padded LDS row stride (floats): banks = (4r + c) mod 64 -> conflict-free

typedef __attribute__((ext_vector_type(2))) float v2f;
typedef __attribute__((ext_vector_type(8))) float v8f;

__launch_bounds__(256, 1)
__global__ void crf_fwd_kernel(const float* __restrict__ P,
                               const float* __restrict__ A,
                               const int*   __restrict__ Y,
                               const int*   __restrict__ L,
                               float* __restrict__ out)
{
    __shared__ float S[16 * MP];    // alpha (S_prev), 16 batches x 128 states
    __shared__ float U[16 * MP];    // u = exp(S - m)
    __shared__ float Pt[16 * MP];   // P rows for current step
    __shared__ float red[16 * 16];  // partial reductions
    __shared__ float mrow[16];      // per-row max

    const int tid  = threadIdx.x;
    const int wave = tid >> 5;
    const int lane = tid & 31;
    const int base = blockIdx.x * 16;   // batch base (4 blocks x 16 batches)

    const int r8 = tid >> 4;   // row owned in reduce phases (0..15)
    const int c8 = tid & 15;   // column group within the row
    const int j0 = c8 * 8;

    // ---- one-time: B operand fragments E = exp(A_trans), column tile jt = wave ----
    // B (4x16, K x N) layout: VGPR0 = K rows {0,2}, VGPR1 = K rows {1,3};
    // lanes 0-15 -> K=0/1, lanes 16-31 -> K=2/3; N = lane % 16.
    const int bcol = (wave << 4) + (lane & 15);
    const int koff = (lane >> 4) << 1;  // 0 for lanes 0-15, 2 for lanes 16-31
    v2f Bfrag[32];
#pragma unroll
    for (int kk = 0; kk < 32; ++kk) {
        const int k0 = 4 * kk + koff;
        Bfrag[kk].x = __expf(A[(k0    ) * M_ + bcol]);
        Bfrag[kk].y = __expf(A[(k0 + 1) * M_ + bcol]);
    }

    // ---- init S0 = P[:,0,:] + A_start ----
    {
        const float* Prow = P + (size_t)(base + r8) * T_ * M_;  // t = 0
#pragma unroll
        for (int e = 0; e < 8; ++e) {
            const int j = j0 + e;
            S[r8 * MP + j] = Prow[j] + A[M_ * M_ + j];          // A_start = A[128,:]
        }
    }

    // ---- per-row bookkeeping state (threads 0..15 persist it in registers) ----
    float sY = 0.f; int yprev = 0, Lb = -1;
    if (tid < 16) {
        const int b = base + tid;
        Lb    = L[b];
        yprev = Y[(size_t)b * T_];                               // y_0
        sY    = P[(size_t)b * T_ * M_ + yprev] + A[M_ * M_ + yprev];
    }
    __syncthreads();

    for (int t = 1; t < T_; ++t) {
        // load P_t rows (coalesced 512B per row), prefetch next step
        {
            const float* Prow = P + (size_t)(base + r8) * T_ * M_ + (size_t)t * M_;
#pragma unroll
            for (int e = 0; e < 8; ++e)
                Pt[r8 * MP + j0 + e] = Prow[j0 + e];
            if (t + 1 < T_)
                __builtin_prefetch(Prow + M_ + j0, 0, 0);        // global_prefetch_b8
        }
        // partial row max of S_prev
        {
            float pm = S[r8 * MP + j0];
#pragma unroll
            for (int e = 1; e < 8; ++e) pm = fmaxf(pm, S[r8 * MP + j0 + e]);
            red[r8 * 16 + c8] = pm;
        }
        __syncthreads();
        float m = red[r8 * 16];
#pragma unroll
        for (int c = 1; c < 16; ++c) m = fmaxf(m, red[r8 * 16 + c]);
        if (c8 == 0) mrow[r8] = m;
        __syncthreads();   // all max reads of red done before sum writes
        // u = exp(S - m), partial row sums
        {
            float ps = 0.f;
#pragma unroll
            for (int e = 0; e < 8; ++e) {
                const float x = __expf(S[r8 * MP + j0 + e] - m);
                U[r8 * MP + j0 + e] = x;
                ps += x;
            }
            red[r8 * 16 + c8] = ps;
        }
        __syncthreads();   // U, red(sums), mrow, Pt visible to all

        // bookkeeping: loss for k = t-1 (logsumexp(S_{t-1}) is m + log(sum u)), then advance sY
        if (tid < 16) {
            float su = 0.f;
#pragma unroll
            for (int c = 0; c < 16; ++c) su += red[tid * 16 + c];
            if (Lb - 1 == t - 1)
                out[base + tid] = mrow[tid] + __logf(su) - sY;
            const int yt = Y[(size_t)(base + tid) * T_ + t];
            sY += Pt[tid * MP + yt] + A[yprev * M_ + yt];
            yprev = yt;
        }

        // GEMM: C(16x16) = U(16x128) * E(128x16) via fp32 WMMA, K split into 32 k-tiles
        // A (16x4) layout: lanes 0-15 -> K={4kk,4kk+1}, lanes 16-31 -> K={4kk+2,4kk+3}; M = lane%16.
        v8f acc = {};
#pragma unroll
        for (int kk = 0; kk < 32; ++kk) {
            const int k0 = 4 * kk + koff;
            v2f a;
            a.x = U[(lane & 15) * MP + k0];
            a.y = U[(lane & 15) * MP + k0 + 1];
            acc = __builtin_amdgcn_wmma_f32_16x16x4_f32(
                      false, a, false, Bfrag[kk], (short)0, acc, false, false);
        }
        // S_new = m_r + log(acc) + P_t ; C layout: VGPR v -> M = v + (lane<16 ? 0 : 8), N = lane%16
        {
            const int j  = (wave << 4) + (lane & 15);
            const int rb = (lane >> 4) << 3;
#pragma unroll
            for (int v = 0; v < 8; ++v) {
                const int r = v + rb;
                S[r * MP + j] = mrow[r] + __logf(acc[v]) + Pt[r * MP + j];
            }
        }
        __syncthreads();   // S_t complete before next iteration reads it
    }

    // ---- epilogue: loss for k = T-1 ----
    {
        float pm = S[r8 * MP + j0];
#pragma unroll
        for (int e = 1; e < 8; ++e) pm = fmaxf(pm, S[r8 * MP + j0 + e]);
        red[r8 * 16 + c8] = pm;
    }
    __syncthreads();
    float m = red[r8 * 16];
#pragma unroll
    for (int c = 1; c < 16; ++c) m = fmaxf(m, red[r8 * 16 + c]);
    if (c8 == 0) mrow[r8] = m;
    __syncthreads();
    {
        float ps = 0.f;
#pragma unroll
        for (int e = 0; e < 8; ++e) ps += __expf(S[r8 * MP + j0 + e] - m);
        red[r8 * 16 + c8] = ps;
    }
    __syncthreads();
    if (tid < 16 && Lb == T_) {
        float su = 0.f;
#pragma unroll
        for (int c = 0; c < 16; ++c) su += red[tid * 16 + c];
        out[base + tid] = mrow[tid] + __logf(su) - sY;
    }
}

extern "C" void kernel_launch(void* const* d_in, const int* in_sizes, int n_in,
                              void* d_out, int out_size, void* d_ws, size_t ws_size,
                              hipStream_t stream) {
    (void)in_sizes; (void)n_in; (void)d_ws; (void)ws_size; (void)out_size;
    const float* P = (const float*)d_in[0];
    const float* A = (const float*)d_in[1];
    const int*   Y = (const int*)d_in[2];
    const int*   L = (const int*)d_in[3];
    float* out = (float*)d_out;
    crf_fwd_kernel<<<B_ / 16, 256, 0, stream>>>(P, A, Y, L, out);
}